// WiseConvolution_13391708029564
// MI455X (gfx1250) — compile-verified
//
#include <hip/hip_runtime.h>
#include <hip/hip_bf16.h>

// Problem constants (match the reference).
#define NROWS 500000
#define NSEG  10000
#define DIM   128   // D_IN == D_OUT == 128

typedef __attribute__((ext_vector_type(2))) float v2f;
typedef __attribute__((ext_vector_type(8))) float v8f;

// ---------------------------------------------------------------------------
// Phase 1: contiguous segment reduction of input rows.
//   segsum[s][c] = sum_{r=start[s]..end[s]} input[r][c]
// One block (128 threads = 4 waves) per segment.
//   thread t: row-group rg = t>>5 (0..3), col-group cg = t&31 (float4 columns)
// Each iteration the block reads 4 rows x 512B = 2KB, fully coalesced b128s.
// Total traffic: 256 MB read once  ->  ~11 us at 23.3 TB/s (the roofline).
// ---------------------------------------------------------------------------
__global__ __launch_bounds__(128)
void seg_reduce_kernel(const float* __restrict__ in,
                       const long long* __restrict__ graph,
                       float* __restrict__ segsum) {
    const int s = blockIdx.x;
    const long long start = graph[2 * s];
    const long long end   = graph[2 * s + 1];   // inclusive

    const int t  = threadIdx.x;
    const int rg = t >> 5;   // 0..3  row group
    const int cg = t & 31;   // 0..31 float4 column group

    float4 acc = make_float4(0.f, 0.f, 0.f, 0.f);
    const float4* __restrict__ base = (const float4*)in;  // row stride = 32 float4
    for (long long r = start + rg; r <= end; r += 4) {
        float4 v = base[r * 32 + cg];
        acc.x += v.x; acc.y += v.y; acc.z += v.z; acc.w += v.w;
    }

    __shared__ float4 lds[128];
    lds[t] = acc;
    __syncthreads();

    if (rg == 0) {
        float4 a0 = lds[cg];
        float4 a1 = lds[32 + cg];
        float4 a2 = lds[64 + cg];
        float4 a3 = lds[96 + cg];
        float4 r;
        r.x = (a0.x + a1.x) + (a2.x + a3.x);
        r.y = (a0.y + a1.y) + (a2.y + a3.y);
        r.z = (a0.z + a1.z) + (a2.z + a3.z);
        r.w = (a0.w + a1.w) + (a2.w + a3.w);
        ((float4*)segsum)[s * 32 + cg] = r;
    }
}

// ---------------------------------------------------------------------------
// Phase 2: out = segsum @ W^T + count*b  via V_WMMA_F32_16X16X4_F32.
// One wave (32 threads) per 16x16 output tile; grid = (625 M-tiles, 8 N-tiles).
// K loop: 128 / 4 = 32 WMMAs per wave.
//
// Fragment layouts (CDNA5 ISA 7.12.2, wave32):
//   A 16x4 f32, 2 VGPRs:  lane = half*16 + m;  v0 = A[m][k+2*half], v1 = A[m][k+2*half+1]
//   B 4x16 f32, 2 VGPRs:  lane = half*16 + n;  v0 = B[k+2*half][n], v1 = B[k+2*half+1][n]
//   C/D 16x16 f32, 8 VGPRs: c[r] = D[r + 8*half][n],  n = lane & 15
// out[s][j] = sum_k segsum[s][k] * W[j][k]  (B[k][n] = W[tn+n][k])
// ---------------------------------------------------------------------------
__global__ __launch_bounds__(32)
void seg_gemm_wmma_kernel(const float* __restrict__ segsum,
                          const long long* __restrict__ graph,
                          const float* __restrict__ W,
                          const float* __restrict__ bias,
                          float* __restrict__ out) {
    const int tm   = blockIdx.x << 4;   // segment tile base (M)
    const int tn   = blockIdx.y << 4;   // output-column tile base (N)
    const int lane = threadIdx.x;       // 0..31
    const int half = lane >> 4;         // 0 | 1
    const int ml   = lane & 15;         // M index for A, N index for B/C

    const float* __restrict__ arow = segsum + (size_t)(tm + ml) * DIM + half * 2;
    const float* __restrict__ brow = W      + (size_t)(tn + ml) * DIM + half * 2;

    v8f c = {};  // zero accumulator
    #pragma unroll
    for (int k = 0; k < DIM; k += 4) {
        v2f a, b;
        a.x = arow[k]; a.y = arow[k + 1];
        b.x = brow[k]; b.y = brow[k + 1];
        // 8 args: (neg_a, A, neg_b, B, c_mod, C, reuse_a, reuse_b)
        c = __builtin_amdgcn_wmma_f32_16x16x4_f32(
                /*neg_a=*/false, a, /*neg_b=*/false, b,
                /*c_mod=*/(short)0, c, /*reuse_a=*/false, /*reuse_b=*/false);
    }

    // Fold in count[s] * b[j], then store the tile.
    const float bj = bias[tn + ml];
    const int sbase = tm + half * 8;
    #pragma unroll
    for (int r = 0; r < 8; ++r) {
        const int s = sbase + r;
        const long long cnt = graph[2 * s + 1] - graph[2 * s] + 1;
        out[(size_t)s * DIM + tn + ml] = c[r] + (float)cnt * bj;
    }
}

// ---------------------------------------------------------------------------
// Launch: inputs in setup_inputs() order: input(f32), graph(i64), W(f32), b(f32)
// d_ws: needs NSEG*DIM*4 = 5.12 MB for segment sums.
// ---------------------------------------------------------------------------
extern "C" void kernel_launch(void* const* d_in, const int* in_sizes, int n_in,
                              void* d_out, int out_size, void* d_ws, size_t ws_size,
                              hipStream_t stream) {
    const float*     input = (const float*)d_in[0];
    const long long* graph = (const long long*)d_in[1];
    const float*     W     = (const float*)d_in[2];
    const float*     b     = (const float*)d_in[3];
    float*           out   = (float*)d_out;
    float*           segsum = (float*)d_ws;   // NSEG * DIM floats

    // Phase 1: one block per segment; 128 threads (4 waves).
    seg_reduce_kernel<<<NSEG, 128, 0, stream>>>(input, graph, segsum);

    // Phase 2: one wave per 16x16 tile; 625 x 8 tiles.
    dim3 grid2(NSEG / 16, DIM / 16);
    seg_gemm_wmma_kernel<<<grid2, 32, 0, stream>>>(segsum, graph, W, b, out);
}